// FCOSOutputs3D_16784732192913
// MI455X (gfx1250) — compile-verified
//
#include <hip/hip_runtime.h>
#include <hip/hip_bf16.h>
#include <stdint.h>

#define INF_VAL 100000000.0f
#define NUM_CLASSES 2
#define MAX_G 64   // actual G = 32; LDS sized generously

typedef __attribute__((address_space(1))) int g_int;
typedef __attribute__((address_space(3))) int l_int;

// ---------------------------------------------------------------------------
// CDNA5 async global->LDS copy (ASYNCcnt path). Guarded so the file compiles
// on toolchains without the builtin; fallback is a plain LDS store.
// ---------------------------------------------------------------------------
__device__ __forceinline__ void async_copy_b32_to_lds(const void* gsrc, void* lds_dst) {
#if __has_builtin(__builtin_amdgcn_global_load_async_to_lds_b32)
    __builtin_amdgcn_global_load_async_to_lds_b32(
        (g_int*)(uintptr_t)gsrc,
        (l_int*)(uint32_t)(uintptr_t)lds_dst,
        0, 0);
#else
    *(volatile uint32_t*)lds_dst = *(const uint32_t*)gsrc;
#endif
}

__device__ __forceinline__ void wait_async_copies() {
#if __has_builtin(__builtin_amdgcn_s_wait_asynccnt)
    __builtin_amdgcn_s_wait_asynccnt(0);
#else
    asm volatile("s_wait_asynccnt 0" ::: "memory");
#endif
}

// ---------------------------------------------------------------------------
// One thread = one (image b, location p). Box table staged in LDS per block.
// ---------------------------------------------------------------------------
__global__ __launch_bounds__(256)
void fcos3d_assign_kernel(const float* __restrict__ locations,    // (N,3)
                          const float* __restrict__ size_ranges,  // (N,2)
                          const float* __restrict__ loc_strides,  // (N,)
                          const float* __restrict__ gt_boxes,     // (B,G,6)
                          const int*   __restrict__ gt_classes,   // (B,G)
                          float* __restrict__ out_labels,         // (B,N)
                          float* __restrict__ out_reg,            // (B,N,6)
                          float* __restrict__ out_tinds,          // (B,N)
                          int N, int G)
{
    __shared__ float s_box[MAX_G * 6];
    __shared__ float s_area[MAX_G];
    __shared__ int   s_cls[MAX_G];

    const int b = blockIdx.y;
    const int t = threadIdx.x;
    const int nbox_dw = G * 6;

    // Stage this image's box table + classes into LDS via the async path.
    if (t < nbox_dw)
        async_copy_b32_to_lds(&gt_boxes[(size_t)b * nbox_dw + t], &s_box[t]);
    if (t < G)
        async_copy_b32_to_lds(&gt_classes[(size_t)b * G + t], &s_cls[t]);
    wait_async_copies();          // each wave drains its own ASYNCcnt
    __syncthreads();              // then all waves see the completed LDS

    // Precompute box volumes once per block.
    if (t < G) {
        const float* bx = &s_box[t * 6];
        s_area[t] = (bx[3] - bx[0]) * (bx[4] - bx[1]) * (bx[5] - bx[2]);
    }
    __syncthreads();

    const int p = blockIdx.x * blockDim.x + t;
    if (p >= N) return;

    const float x   = locations[3 * p + 0];
    const float y   = locations[3 * p + 1];
    const float z   = locations[3 * p + 2];
    const float sr0 = size_ranges[2 * p + 0];
    const float sr1 = size_ranges[2 * p + 1];

    float best  = INF_VAL;   // all-INF => bestj stays 0, matching jnp.argmin
    int   bestj = 0;

#pragma unroll 8
    for (int j = 0; j < G; ++j) {
        // Uniform LDS address across the wave -> broadcast, no bank conflicts.
        const float b0 = s_box[6 * j + 0];
        const float b1 = s_box[6 * j + 1];
        const float b2 = s_box[6 * j + 2];
        const float b3 = s_box[6 * j + 3];
        const float b4 = s_box[6 * j + 4];
        const float b5 = s_box[6 * j + 5];

        const float l  = x  - b0;
        const float tt = y  - b1;
        const float nn = z  - b2;
        const float r  = b3 - x;
        const float bb = b4 - y;
        const float f  = b5 - z;

        const float mn = fminf(fminf(fminf(l, tt), fminf(nn, r)), fminf(bb, f));
        const float mx = fmaxf(fmaxf(fmaxf(l, tt), fmaxf(nn, r)), fmaxf(bb, f));

        const int ok = (mn > 0.0f) & (mx >= sr0) & (mx <= sr1);
        const float a = ok ? s_area[j] : INF_VAL;
        if (a < best) { best = a; bestj = j; }   // strict '<' => first-hit argmin
    }

    // Gather winning box, rebuild reg, scale by 1/stride (exact division).
    const float b0 = s_box[6 * bestj + 0];
    const float b1 = s_box[6 * bestj + 1];
    const float b2 = s_box[6 * bestj + 2];
    const float b3 = s_box[6 * bestj + 3];
    const float b4 = s_box[6 * bestj + 4];
    const float b5 = s_box[6 * bestj + 5];
    const float st = loc_strides[p];

    const size_t base = (size_t)b * N + p;
    out_labels[base] = (best == INF_VAL) ? (float)NUM_CLASSES : (float)s_cls[bestj];

    float* rp = out_reg + base * 6;
    rp[0] = (x  - b0) / st;
    rp[1] = (y  - b1) / st;
    rp[2] = (z  - b2) / st;
    rp[3] = (b3 - x)  / st;
    rp[4] = (b4 - y)  / st;
    rp[5] = (b5 - z)  / st;

    out_tinds[base] = (float)(bestj + b * G);
}

extern "C" void kernel_launch(void* const* d_in, const int* in_sizes, int n_in,
                              void* d_out, int out_size, void* d_ws, size_t ws_size,
                              hipStream_t stream) {
    const float* locations   = (const float*)d_in[0];
    const float* size_ranges = (const float*)d_in[1];
    const float* loc_strides = (const float*)d_in[2];
    const float* gt_boxes    = (const float*)d_in[3];
    const int*   gt_classes  = (const int*)d_in[4];

    const int N  = in_sizes[2];                 // locations count
    const int BG = in_sizes[4];                 // B*G
    const int B  = out_size / (N * 8);          // out = B*N*(1 + 6 + 1)
    const int G  = BG / B;

    float* out        = (float*)d_out;
    float* out_labels = out;                                  // B*N
    float* out_reg    = out + (size_t)B * N;                  // B*N*6
    float* out_tinds  = out + (size_t)B * N * 7;              // B*N

    dim3 grid((N + 255) / 256, B);
    fcos3d_assign_kernel<<<grid, 256, 0, stream>>>(
        locations, size_ranges, loc_strides, gt_boxes, gt_classes,
        out_labels, out_reg, out_tinds, N, G);
}